// ProposalLayer_10505490006034
// MI455X (gfx1250) — compile-verified
//
#include <hip/hip_runtime.h>
#include <cmath>

// ---------------- problem constants (match reference) ----------------
#define BATCH   8
#define NPTS    16384        // power of two -> clean bitonic sort
#define REGCH   76
#define K_PRE   9000
#define K_POST  100
#define NMS_THR 0.8f

#define TWO_PI_F 6.28318530717958647692f
#define PI_F     3.14159265358979323846f
#define ANCH0 1.52563191462f
#define ANCH1 1.62856739989f
#define ANCH2 3.88311640418f

typedef unsigned int u32x4 __attribute__((ext_vector_type(4)));
typedef int          i32x4 __attribute__((ext_vector_type(4)));
typedef int          i32x8 __attribute__((ext_vector_type(8)));

// address-space qualified element types for the async-to-LDS builtins,
// whose params are (AS1 int*, AS3 int*, imm, imm) per the clang diag.
typedef __attribute__((address_space(1))) int   gi32;
typedef __attribute__((address_space(3))) int   li32;
typedef __attribute__((address_space(1))) i32x4 gi32x4;
typedef __attribute__((address_space(3))) i32x4 li32x4;
typedef __attribute__((address_space(3))) void  lvoid_t;

#if __has_builtin(__builtin_amdgcn_global_load_async_to_lds_b32)
#define HAVE_ASYNC32 1
#endif
#if __has_builtin(__builtin_amdgcn_global_load_async_to_lds_b128)
#define HAVE_ASYNC128 1
#endif
#if __has_builtin(__builtin_amdgcn_tensor_load_to_lds)
#define HAVE_TDM 1
#endif

__device__ __forceinline__ void wait_async_zero() {
#if __has_builtin(__builtin_amdgcn_s_wait_asynccnt)
  __builtin_amdgcn_s_wait_asynccnt(0);
#else
  asm volatile("s_wait_asynccnt 0" ::: "memory");
#endif
}

// =====================================================================
// Kernel 1: decode_bbox_target (+ y += h/2).  One thread per point.
// Memory-bound: 8*16384*76*4B ~= 40MB, ~1.7us at 23.3 TB/s.
// =====================================================================
__global__ void decode_kernel(const float* __restrict__ reg,
                              const float* __restrict__ xyz,
                              float* __restrict__ boxes) {
  int p = blockIdx.x * blockDim.x + threadIdx.x;       // 0 .. B*N-1
  if (p >= BATCH * NPTS) return;
  const float* r = reg + (size_t)p * REGCH;
  // prefetch the next row this wave will want (gfx1250 global_prefetch_b8)
  __builtin_prefetch(r + REGCH, 0, 0);

  float bx = -3.402823e38f, bz = -3.402823e38f;
  int xb = 0, zb = 0;
  #pragma unroll
  for (int k = 0; k < 12; ++k) {
    float vx = r[k];       if (vx > bx) { bx = vx; xb = k; }
    float vz = r[12 + k];  if (vz > bz) { bz = vz; zb = k; }
  }
  float pos_x = (float)xb * 0.5f + 0.25f - 3.0f + r[24 + xb] * 0.5f;
  float pos_z = (float)zb * 0.5f + 0.25f - 3.0f + r[36 + zb] * 0.5f;

  float yv = xyz[(size_t)p * 3 + 1] + r[48];

  float bh = -3.402823e38f; int hb = 0;
  #pragma unroll
  for (int k = 0; k < 12; ++k) {
    float v = r[49 + k]; if (v > bh) { bh = v; hb = k; }
  }
  float res = r[61 + hb];
  const float ang = TWO_PI_F / 12.0f;
  float ry = fmodf((float)hb * ang + res * (ang * 0.5f), TWO_PI_F);
  if (ry < 0.0f) ry += TWO_PI_F;       // python % is non-negative
  if (ry > PI_F) ry -= TWO_PI_F;

  float h = r[73] * ANCH0 + ANCH0;
  float w = r[74] * ANCH1 + ANCH1;
  float l = r[75] * ANCH2 + ANCH2;

  float* o = boxes + (size_t)p * 7;
  o[0] = pos_x + xyz[(size_t)p * 3 + 0];
  o[1] = yv + h * 0.5f;                 // props[:,1] += h/2
  o[2] = pos_z + xyz[(size_t)p * 3 + 2];
  o[3] = h; o[4] = w; o[5] = l; o[6] = ry;
}

// =====================================================================
// Kernel 2: per-batch descending bitonic sort of (score, idx) fully in
// LDS. 16384 pairs = 128KB, fits a 320KB WGP. One workgroup per batch.
// Scores staged with GLOBAL_LOAD_ASYNC_TO_LDS when available.
// =====================================================================
__global__ void sort_kernel(const float* __restrict__ scores,
                            float* __restrict__ top_s,
                            int* __restrict__ top_i) {
  extern __shared__ float smem[];
  float* keys = smem;                 // NPTS floats
  int*   vals = (int*)(smem + NPTS);  // NPTS ints
  const int b   = blockIdx.x;
  const int tid = threadIdx.x;
  const int bs  = blockDim.x;
  const float* sb = scores + (size_t)b * NPTS;

  for (int i = tid; i < NPTS; i += bs) {
#if defined(HAVE_ASYNC32)
    __builtin_amdgcn_global_load_async_to_lds_b32(
        (gi32*)(sb + i), (li32*)(keys + i), 0, 0);
#else
    keys[i] = sb[i];
#endif
    vals[i] = i;
  }
#if defined(HAVE_ASYNC32)
  wait_async_zero();
#endif
  __syncthreads();

  // descending bitonic sort
  for (unsigned k = 2; k <= NPTS; k <<= 1) {
    for (unsigned j = k >> 1; j > 0; j >>= 1) {
      for (unsigned t = tid; t < NPTS / 2; t += bs) {
        unsigned i = ((t & ~(j - 1u)) << 1) | (t & (j - 1u));
        unsigned p = i | j;
        bool dsc = ((i & k) == 0);          // this run sorts descending
        float ki = keys[i], kp = keys[p];
        if ((ki < kp) == dsc) {
          keys[i] = kp; keys[p] = ki;
          int v = vals[i]; vals[i] = vals[p]; vals[p] = v;
        }
      }
      __syncthreads();
    }
  }

  for (int i = tid; i < K_PRE; i += bs) {
    top_s[(size_t)b * K_PRE + i] = keys[i];
    top_i[(size_t)b * K_PRE + i] = vals[i];
  }
}

// =====================================================================
// Kernel 3: build BEV planes (SoA: x1,y1,x2,y2,area) for the top-9000.
// =====================================================================
__global__ void bev_kernel(const float* __restrict__ boxes,
                           const int* __restrict__ top_i,
                           float* __restrict__ bev) {
  int g = blockIdx.x * blockDim.x + threadIdx.x;
  if (g >= BATCH * K_PRE) return;
  int b = g / K_PRE, i = g % K_PRE;
  int gi = top_i[g];
  const float* bx = boxes + ((size_t)b * NPTS + gi) * 7;
  float x = bx[0], z = bx[2], w = bx[4], l = bx[5];
  float hl = l * 0.5f, hw = w * 0.5f;
  float x1 = x - hl, y1 = z - hw, x2 = x + hl, y2 = z + hw;
  float* base = bev + (size_t)b * 5 * K_PRE;
  base[0 * K_PRE + i] = x1;
  base[1 * K_PRE + i] = y1;
  base[2 * K_PRE + i] = x2;
  base[3 * K_PRE + i] = y2;
  base[4 * K_PRE + i] = (x2 - x1) * (y2 - y1);
}

// =====================================================================
// Kernel 4: greedy NMS entirely in LDS (180KB tile staged by the TDM),
// then compaction of the first 100 survivors. One workgroup per batch.
// =====================================================================
__global__ void nms_kernel(const float* __restrict__ bev,
                           const float* __restrict__ boxes,
                           const float* __restrict__ top_s,
                           const int* __restrict__ top_i,
                           float* __restrict__ out) {
  extern __shared__ float sm[];
  float* x1 = sm;
  float* y1 = sm + 1 * K_PRE;
  float* x2 = sm + 2 * K_PRE;
  float* y2 = sm + 3 * K_PRE;
  float* ar = sm + 4 * K_PRE;
  int* keep = (int*)(sm + 5 * K_PRE);

  const int b   = blockIdx.x;
  const int tid = threadIdx.x;
  const int bs  = blockDim.x;
  const float* src = bev + (size_t)b * 5 * K_PRE;   // 45000 contiguous dwords

  for (int i = tid; i < K_PRE; i += bs) keep[i] = 1;

#if defined(HAVE_TDM)
  if (tid == 0) {
    // Tensor DMA descriptor (CDNA5 ISA): 1-D tile,
    // 45000 x 4B elements, global->LDS.
    unsigned long long ga = (unsigned long long)src;
    unsigned lds_addr = (unsigned)(unsigned long long)(lvoid_t*)sm;
    const unsigned E = 5u * K_PRE;                  // 45000 < 2^16
    u32x4 g0;
    g0[0] = 1u;                                     // count=1, user mode
    g0[1] = lds_addr;                               // lds_addr [63:32]
    g0[2] = (unsigned)(ga & 0xFFFFFFFFull);         // global_addr lo
    g0[3] = ((unsigned)(ga >> 32) & 0x01FFFFFFu) | (2u << 30); // addr hi | type=2
    i32x8 g1;
    g1[0] = (int)(2u << 16);                        // data_size = 4B
    g1[1] = (int)((E & 0xFFFFu) << 16);             // tensor_dim0 lo16 @bit48
    g1[2] = (int)((E >> 16) | (1u << 16));          // tensor_dim0 hi | tensor_dim1=1
    g1[3] = (int)((E & 0xFFFFu) << 16);             // tile_dim0 = E
    g1[4] = 1;                                      // tile_dim1 = 1, tile_dim2 = 0
    g1[5] = (int)E;                                 // tensor_dim0_stride lo32
    g1[6] = 0;
    g1[7] = 0;
    i32x4 z4 = {0, 0, 0, 0};
#if __clang_major__ >= 23
    i32x8 z8 = {0, 0, 0, 0, 0, 0, 0, 0};
    __builtin_amdgcn_tensor_load_to_lds(g0, g1, z4, z4, z8, 0);
#else
    __builtin_amdgcn_tensor_load_to_lds(g0, g1, z4, z4, 0);
#endif
    __builtin_amdgcn_s_wait_tensorcnt(0);
  }
#elif defined(HAVE_ASYNC128)
  for (int i4 = tid * 4; i4 < 5 * K_PRE; i4 += bs * 4) {   // 45000 % 4 == 0
    __builtin_amdgcn_global_load_async_to_lds_b128(
        (gi32x4*)(src + i4), (li32x4*)(sm + i4), 0, 0);
  }
  wait_async_zero();
#else
  for (int i = tid; i < 5 * K_PRE; i += bs) sm[i] = src[i];
#endif
  __syncthreads();

  // greedy sequential NMS: iteration i suppresses j>i; matches the
  // reference fori_loop exactly (suppressed i never suppress others).
  for (int i = 0; i < K_PRE; ++i) {
    if (keep[i]) {
      float xi1 = x1[i], yi1 = y1[i], xi2 = x2[i], yi2 = y2[i], ai = ar[i];
      for (int j = i + 1 + tid; j < K_PRE; j += bs) {
        if (keep[j]) {
          float xx1 = fmaxf(xi1, x1[j]);
          float yy1 = fmaxf(yi1, y1[j]);
          float xx2 = fminf(xi2, x2[j]);
          float yy2 = fminf(yi2, y2[j]);
          float inter = fmaxf(xx2 - xx1, 0.0f) * fmaxf(yy2 - yy1, 0.0f);
          float iou = inter / (ai + ar[j] - inter);
          if (iou > NMS_THR) keep[j] = 0;
        }
      }
    }
    __syncthreads();
  }

  // zero-fill this batch's outputs, then serial compaction by thread 0
  float* out_b = out;                               // (B,100,7)
  float* out_s = out + (size_t)BATCH * K_POST * 7;  // (B,100)
  for (int q = tid; q < K_POST * 7; q += bs) out_b[(size_t)b * K_POST * 7 + q] = 0.0f;
  for (int q = tid; q < K_POST; q += bs)     out_s[(size_t)b * K_POST + q] = 0.0f;
  __syncthreads();
  if (tid == 0) {
    int c = 0;
    for (int i = 0; i < K_PRE && c < K_POST; ++i) {
      if (keep[i]) {
        int gi = top_i[(size_t)b * K_PRE + i];
        const float* s7 = boxes + ((size_t)b * NPTS + gi) * 7;
        float* d7 = out_b + ((size_t)b * K_POST + c) * 7;
        #pragma unroll
        for (int k = 0; k < 7; ++k) d7[k] = s7[k];
        out_s[(size_t)b * K_POST + c] = top_s[(size_t)b * K_PRE + i];
        ++c;
      }
    }
  }
}

// =====================================================================
extern "C" void kernel_launch(void* const* d_in, const int* in_sizes, int n_in,
                              void* d_out, int out_size, void* d_ws, size_t ws_size,
                              hipStream_t stream) {
  const float* rpn_scores = (const float*)d_in[0];   // (B,N)
  const float* rpn_reg    = (const float*)d_in[1];   // (B,N,76)
  const float* xyz        = (const float*)d_in[2];   // (B,N,3)
  float* out = (float*)d_out;                        // 5600 boxes + 800 scores

  // workspace layout (~5.7 MB)
  float* boxes = (float*)d_ws;                                   // B*N*7
  float* top_s = boxes + (size_t)BATCH * NPTS * 7;               // B*9000
  int*   top_i = (int*)(top_s + (size_t)BATCH * K_PRE);          // B*9000
  float* bev   = (float*)(top_i + (size_t)BATCH * K_PRE);        // B*9000*5

  decode_kernel<<<(BATCH * NPTS + 255) / 256, 256, 0, stream>>>(rpn_reg, xyz, boxes);
  sort_kernel<<<BATCH, 1024, NPTS * (sizeof(float) + sizeof(int)), stream>>>(
      rpn_scores, top_s, top_i);
  bev_kernel<<<(BATCH * K_PRE + 255) / 256, 256, 0, stream>>>(boxes, top_i, bev);
  nms_kernel<<<BATCH, 512, (5 * K_PRE) * sizeof(float) + K_PRE * sizeof(int), stream>>>(
      bev, boxes, top_s, top_i, out);
}